// SetAbstractionMSG_8830452761364
// MI455X (gfx1250) — compile-verified
//
#include <hip/hip_runtime.h>

typedef __attribute__((ext_vector_type(2))) float v2f;
typedef __attribute__((ext_vector_type(8))) float v8f;

// ---------------------------------------------------------------------------
// FPS: one block per batch. xyz cached in LDS, argmax via wave shuffle
// butterfly + cross-wave LDS reduce, tie-break = smallest index (jnp.argmax).
// Writes new_xyz (B,1024,3) into d_out.
// ---------------------------------------------------------------------------
__global__ __launch_bounds__(1024) void fps_kernel(const float* __restrict__ xyz,
                                                   float* __restrict__ newxyz) {
  __shared__ float px[4096], py[4096], pz[4096];
  __shared__ float rD[32];
  __shared__ int rI[32];
  __shared__ int cid[1024];
  __shared__ int s_far;
  const int b = blockIdx.x;
  const int tid = threadIdx.x;
  const float* base = xyz + (size_t)b * 4096 * 3;
  for (int i = tid; i < 4096; i += 1024) {
    px[i] = base[i * 3 + 0];
    py[i] = base[i * 3 + 1];
    pz[i] = base[i * 3 + 2];
  }
  float dist[4];
  dist[0] = dist[1] = dist[2] = dist[3] = 3.4e38f;
  if (tid == 0) { s_far = 0; cid[0] = 0; }
  __syncthreads();
  const int lane = tid & 31, wave = tid >> 5;
  for (int t = 1; t < 1024; ++t) {
    const int far = s_far;
    const float cx = px[far], cy = py[far], cz = pz[far];
    float bd = -1.0f;
    int bi = 0;
#pragma unroll
    for (int j = 0; j < 4; ++j) {
      const int p = tid * 4 + j;
      const float dx = px[p] - cx, dy = py[p] - cy, dz = pz[p] - cz;
      const float d = dx * dx + dy * dy + dz * dz;
      if (d < dist[j]) dist[j] = d;
      if (dist[j] > bd) { bd = dist[j]; bi = p; }  // strict >: first max
    }
#pragma unroll
    for (int off = 16; off > 0; off >>= 1) {
      const float od = __shfl_xor(bd, off);
      const int oi = __shfl_xor(bi, off);
      if (od > bd || (od == bd && oi < bi)) { bd = od; bi = oi; }
    }
    if (lane == 0) { rD[wave] = bd; rI[wave] = bi; }
    __syncthreads();
    if (wave == 0) {
      bd = rD[lane];
      bi = rI[lane];
#pragma unroll
      for (int off = 16; off > 0; off >>= 1) {
        const float od = __shfl_xor(bd, off);
        const int oi = __shfl_xor(bi, off);
        if (od > bd || (od == bd && oi < bi)) { bd = od; bi = oi; }
      }
      if (lane == 0) { s_far = bi; cid[t] = bi; }
    }
    __syncthreads();
  }
  {
    const int i = cid[tid];
    float* o = newxyz + ((size_t)b * 1024 + tid) * 3;
    o[0] = px[i];
    o[1] = py[i];
    o[2] = pz[i];
  }
}

// ---------------------------------------------------------------------------
// Ball query: one wave32 per query point. Scans the 4096 points in index
// order, emits first k in-radius indices (ascending) via ballot+prefix-popc,
// pads with the first hit (matches reference top_k(-idx) + where(first)).
// ---------------------------------------------------------------------------
__global__ __launch_bounds__(256) void ballquery_kernel(const float* __restrict__ xyz,
                                                        const float* __restrict__ newxyz,
                                                        int* __restrict__ gidx,
                                                        float r2, int k) {
  const int gt = blockIdx.x * 256 + threadIdx.x;
  const int q = gt >> 5;
  const int lane = gt & 31;
  const int b = q >> 10;
  const float cx = newxyz[(size_t)q * 3 + 0];
  const float cy = newxyz[(size_t)q * 3 + 1];
  const float cz = newxyz[(size_t)q * 3 + 2];
  const float* pb = xyz + (size_t)b * 4096 * 3;
  int* out = gidx + (size_t)q * k;
  int found = 0;
  int first = -1;
  for (int basei = 0; basei < 4096; basei += 32) {
    const int p = basei + lane;
    const float dx = pb[p * 3 + 0] - cx;
    const float dy = pb[p * 3 + 1] - cy;
    const float dz = pb[p * 3 + 2] - cz;
    const float d = dx * dx + dy * dy + dz * dz;
    const bool in = (d <= r2);
    const unsigned mm = (unsigned)__ballot(in);
    if (first < 0 && mm != 0u) first = basei + __ffs(mm) - 1;
    const int rank = __popc(mm & ((1u << lane) - 1u));
    const int pos = found + rank;
    if (in && pos < k) out[pos] = p;
    found += __popc(mm);
    if (found >= k) break;
  }
  if (found > k) found = k;
  if (first < 0) first = 0;
  for (int i = found + lane; i < k; i += 32) out[i] = first;
}

// ---------------------------------------------------------------------------
// GEMM:  Y[m,o] = sum_c A[m,c] * W[o,c]   using V_WMMA_F32_16X16X4_F32.
// gatherMode=1: A rows built on the fly from xyz/features/gidx (layer 0,
//               C_in padded 35->36 with zeros).
// gatherMode=0: A = relu(Xin * a + b)   (previous layer's BN+ReLU fused).
// A staged transposed in LDS [K][row] (padded stride), W transposed [K][Cout].
// Also emits deterministic per-block channel sum / sum-of-squares partials.
// Block = 256 thr = 8 waves; wave w -> (rowTile w/nTiles, colTile w%nTiles).
// ---------------------------------------------------------------------------
__global__ __launch_bounds__(256) void gemm_kernel(
    const float* __restrict__ Xin, const float* __restrict__ xyz,
    const float* __restrict__ feat, const float* __restrict__ newxyz,
    const int* __restrict__ gidx, int logk, const float* __restrict__ W,
    const float* __restrict__ ab, float* __restrict__ Y,
    float* __restrict__ Psum, float* __restrict__ Psq, int CinReal, int Cinp,
    int Cout, int nTiles, int RT, int gatherMode) {
  extern __shared__ float smem[];
  const int rows = RT * 16;
  const int RS = rows + 1;  // padded LDS stride (conflict-free A reads)
  float* Xt = smem;                  // [Cinp][RS]
  float* Wt = Xt + Cinp * RS;        // [Cinp][Cout]
  float* wS = Wt + Cinp * Cout;      // [8][16]
  float* wQ = wS + 128;              // [8][16]
  const int tid = threadIdx.x;
  const int rowBase = blockIdx.x * rows;

  // Stage W transposed (pad K column 35 with zeros for gather mode).
  for (int i = tid; i < Cinp * Cout; i += 256) {
    const int cc = i / Cout;
    const int o = i - cc * Cout;
    Wt[i] = (cc < CinReal) ? W[o * CinReal + cc] : 0.0f;
  }
  // Stage A tile transposed, fusing gather or BN+ReLU of previous layer.
  for (int i = tid; i < rows * Cinp; i += 256) {
    const int r = i / Cinp;
    const int c = i - r * Cinp;
    const int m = rowBase + r;
    float v;
    if (gatherMode) {
      if (c < 35) {
        const int idx = gidx[m];
        const int q = m >> logk;
        const int bb = q >> 10;
        if (c < 3)
          v = xyz[((size_t)bb * 4096 + idx) * 3 + c] - newxyz[(size_t)q * 3 + c];
        else
          v = feat[((size_t)bb * 4096 + idx) * 32 + (c - 3)];
      } else {
        v = 0.0f;
      }
    } else {
      const float x = Xin[(size_t)m * CinReal + c];
      v = fmaxf(x * ab[c] + ab[CinReal + c], 0.0f);
    }
    Xt[c * RS + r] = v;
  }
  __syncthreads();

  const int wave = tid >> 5, lane = tid & 31;
  const int nt = wave % nTiles, rt = wave / nTiles;
  const int h = lane >> 4, lr = lane & 15;
  if (rt < RT) {
    v8f acc = {};
    const int rA = rt * 16 + lr;      // A row (M = lane%16)
    const int col = nt * 16 + lr;     // B/D column (N = lane%16)
    for (int k0 = 0; k0 < Cinp; k0 += 4) {
      const int kk = k0 + 2 * h;      // K pair per half-wave
      v2f a, bv;
      a.x = Xt[kk * RS + rA];
      a.y = Xt[(kk + 1) * RS + rA];
      bv.x = Wt[kk * Cout + col];
      bv.y = Wt[(kk + 1) * Cout + col];
      acc = __builtin_amdgcn_wmma_f32_16x16x4_f32(false, a, false, bv,
                                                  (short)0, acc, false, false);
    }
    float s = 0.0f, sq = 0.0f;
#pragma unroll
    for (int v = 0; v < 8; ++v) {
      const int mr = rowBase + rt * 16 + v + 8 * h;  // D row = v + 8*(lane/16)
      const float x = acc[v];
      Y[(size_t)mr * Cout + col] = x;
      s += x;
      sq += x * x;
    }
    s += __shfl_xor(s, 16);
    sq += __shfl_xor(sq, 16);
    if (lane < 16) { wS[wave * 16 + lane] = s; wQ[wave * 16 + lane] = sq; }
  }
  __syncthreads();
  // Deterministic per-block channel partials (fixed combine order over rt).
  for (int i = tid; i < Cout; i += 256) {
    const int nt2 = i >> 4, c2 = i & 15;
    float s = 0.0f, sq = 0.0f;
    for (int rt2 = 0; rt2 < RT; ++rt2) {
      const int w = rt2 * nTiles + nt2;
      s += wS[w * 16 + c2];
      sq += wQ[w * 16 + c2];
    }
    Psum[(size_t)blockIdx.x * Cout + i] = s;
    Psq[(size_t)blockIdx.x * Cout + i] = sq;
  }
}

// ---------------------------------------------------------------------------
// Stats reduce: one block per channel, fixed-order strided sum + LDS tree
// (deterministic). Emits fused BN affine: a = gamma*rsqrt(var+eps),
// b = beta - mean*a.
// ---------------------------------------------------------------------------
__global__ __launch_bounds__(256) void stats_kernel(
    const float* __restrict__ Psum, const float* __restrict__ Psq, int nBlocks,
    int Cout, float Minv, const float* __restrict__ gamma,
    const float* __restrict__ beta, float* __restrict__ ab) {
  __shared__ float sS[256], sQ[256];
  const int ch = blockIdx.x, tid = threadIdx.x;
  float s = 0.0f, q = 0.0f;
  for (int i = tid; i < nBlocks; i += 256) {
    s += Psum[(size_t)i * Cout + ch];
    q += Psq[(size_t)i * Cout + ch];
  }
  sS[tid] = s;
  sQ[tid] = q;
  __syncthreads();
  for (int off = 128; off > 0; off >>= 1) {
    if (tid < off) { sS[tid] += sS[tid + off]; sQ[tid] += sQ[tid + off]; }
    __syncthreads();
  }
  if (tid == 0) {
    const float mean = sS[0] * Minv;
    float var = sQ[0] * Minv - mean * mean;
    var = fmaxf(var, 0.0f);
    const float inv = rsqrtf(var + 1e-5f);
    const float a = gamma[ch] * inv;
    ab[ch] = a;
    ab[Cout + ch] = beta[ch] - mean * a;
  }
}

// ---------------------------------------------------------------------------
// Max-pool over k neighbors with the last layer's BN+ReLU applied on the fly.
// Writes into the per-scale channel slice of new_feat (B,1024,320).
// ---------------------------------------------------------------------------
__global__ __launch_bounds__(256) void maxpool_kernel(
    const float* __restrict__ Y, const float* __restrict__ ab,
    float* __restrict__ out, int k, int Cout, int chOff) {
  const int gt = blockIdx.x * 256 + threadIdx.x;
  const int q = gt / Cout;
  const int o = gt - q * Cout;
  const float a = ab[o], b = ab[Cout + o];
  const float* yp = Y + (size_t)q * k * Cout + o;
  float mx = 0.0f;  // relu outputs are >= 0
  for (int j = 0; j < k; ++j) {
    const float x = yp[(size_t)j * Cout];
    mx = fmaxf(mx, fmaxf(x * a + b, 0.0f));
  }
  out[(size_t)q * 320 + chOff + o] = mx;
}

// ---------------------------------------------------------------------------
extern "C" void kernel_launch(void* const* d_in, const int* in_sizes, int n_in,
                              void* d_out, int out_size, void* d_ws,
                              size_t ws_size, hipStream_t stream) {
  (void)in_sizes; (void)n_in; (void)out_size; (void)ws_size;
  const float* xyz = (const float*)d_in[0];
  const float* feat = (const float*)d_in[1];
  const float *Wp[3][3], *Gp[3][3], *Bp[3][3];
  int t = 2;
  for (int s = 0; s < 3; ++s)
    for (int l = 0; l < 3; ++l) {
      Wp[s][l] = (const float*)d_in[t++];
      Gp[s][l] = (const float*)d_in[t++];
      Bp[s][l] = (const float*)d_in[t++];
    }
  float* out_xyz = (float*)d_out;            // (8,1024,3)
  float* out_feat = out_xyz + 8 * 1024 * 3;  // (8,1024,320)

  // Workspace carve (peak ~502 MB; scales processed sequentially reuse it).
  char* w = (char*)d_ws;
  auto carve = [&](size_t bytes) -> void* {
    void* p = (void*)w;
    w += (bytes + 255) & ~(size_t)255;
    return p;
  };
  int* gidx = (int*)carve((size_t)8192 * 64 * sizeof(int));
  float* ab = (float*)carve(256 * sizeof(float));
  float* Psum = (float*)carve((size_t)32768 * 128 * sizeof(float));
  float* Psq = (float*)carve((size_t)32768 * 128 * sizeof(float));
  float* bufA = (float*)carve((size_t)524288 * 128 * sizeof(float));  // Y1/Y3
  float* bufB = (float*)carve((size_t)524288 * 96 * sizeof(float));   // Y2

  fps_kernel<<<dim3(8), dim3(1024), 0, stream>>>(xyz, out_xyz);

  static const int MLP[3][3] = {{32, 32, 64}, {64, 64, 128}, {64, 96, 128}};
  static const float R2[3] = {0.01f, 0.04f, 0.16f};
  static const int KS[3] = {16, 32, 64};
  static const int LOGK[3] = {4, 5, 6};
  static const int CHOFF[3] = {0, 64, 192};

  for (int s = 0; s < 3; ++s) {
    const int k = KS[s], logk = LOGK[s];
    const int M = 8 * 1024 * k;
    ballquery_kernel<<<dim3(1024), dim3(256), 0, stream>>>(xyz, out_xyz, gidx,
                                                           R2[s], k);
    const int chans[4] = {35, MLP[s][0], MLP[s][1], MLP[s][2]};
    float* Ybuf[3] = {bufA, bufB, bufA};
    const float* Xprev = nullptr;
    for (int l = 0; l < 3; ++l) {
      const int CinReal = chans[l];
      const int Cinp = (l == 0) ? 36 : CinReal;  // pad 35 -> 36 (K%4==0)
      const int Cout = chans[l + 1];
      const int nTiles = Cout / 16;
      int RT = 8 / nTiles;
      if (RT < 1) RT = 1;
      const int rows = RT * 16;
      const int nBlocks = M / rows;
      const size_t smem =
          ((size_t)Cinp * (rows + 1) + (size_t)Cinp * Cout + 256) *
          sizeof(float);
      gemm_kernel<<<dim3(nBlocks), dim3(256), smem, stream>>>(
          Xprev, xyz, feat, out_xyz, gidx, logk, Wp[s][l], ab, Ybuf[l], Psum,
          Psq, CinReal, Cinp, Cout, nTiles, RT, (l == 0) ? 1 : 0);
      stats_kernel<<<dim3(Cout), dim3(256), 0, stream>>>(
          Psum, Psq, nBlocks, Cout, 1.0f / (float)M, Gp[s][l], Bp[s][l], ab);
      Xprev = Ybuf[l];
    }
    maxpool_kernel<<<dim3((8192 * chans[3]) / 256), dim3(256), 0, stream>>>(
        Ybuf[2], ab, out_feat, k, chans[3], CHOFF[s]);
  }
}